// SparseDeConv3dTestTorch_26963804684445
// MI455X (gfx1250) — compile-verified
//
#include <hip/hip_runtime.h>

typedef __attribute__((ext_vector_type(2))) float v2f;
typedef __attribute__((ext_vector_type(8))) float v8f;

// Problem constants (match reference)
#define NB        2
#define CIN       64
#define COUT      64
#define GD        48                 // input grid dim
#define OD        95                 // output dim = (48-1)*2 - 2 + 2 + 1
#define GRID_ELEMS (NB * GD * GD * GD * CIN)     // 14,155,776 f32 = 56.6 MB
#define WB_ELEMS   (27 * 16 * 2 * COUT * 2)      // 110,592 f32 = 442 KB
#define ROW_F32    (17 * CIN)                    // 1088 f32 per staged (z,y) row
#define ROW_F4     (ROW_F32 / 4)                 // 272 float4 per row

// ---------------------------------------------------------------- zero grid
__global__ __launch_bounds__(256) void zero_grid_kernel(float4* __restrict__ g, int n4) {
    int t = blockIdx.x * 256 + threadIdx.x;
    if (t < n4) g[t] = make_float4(0.f, 0.f, 0.f, 0.f);
}

// ------------------------------------------------------------ repack weights
// w1: (CIN, COUT, 3,3,3). WB flat index: (((t*16+cbi)*2+h)*64+co)*2 + j holds
// w1[ci = 4*cbi + 2*h + j][co][tap t], matching the 16x4 f32 A-fragment lane map.
__global__ __launch_bounds__(256) void repack_w_kernel(const float* __restrict__ w1,
                                                       float* __restrict__ wb) {
    int tid = blockIdx.x * 256 + threadIdx.x;
    if (tid >= WB_ELEMS) return;
    int j  = tid & 1;  int r = tid >> 1;
    int co = r & 63;   r >>= 6;
    int h  = r & 1;    r >>= 1;
    int cb = r & 15;   r >>= 4;
    int t  = r;                       // 0..26 = kz*9+ky*3+kx
    int ci = 4 * cb + 2 * h + j;
    wb[tid] = w1[(ci * COUT + co) * 27 + t];
}

// ------------------------------------------------------- scatter features
__global__ __launch_bounds__(256) void scatter_kernel(const float* __restrict__ feat,
                                                      const int* __restrict__ coors,
                                                      float* __restrict__ grid, int total) {
    int tid = blockIdx.x * 256 + threadIdx.x;
    if (tid >= total) return;
    int n = tid >> 6, c = tid & 63;
    int b = coors[n * 4 + 0];
    int z = coors[n * 4 + 1];
    int y = coors[n * 4 + 2];
    int x = coors[n * 4 + 3];
    unsafeAtomicAdd(grid + ((((b * GD + z) * GD + y) * GD + x) * CIN + c), feat[n * CIN + c]);
}

// ---------------------------------------------------- gather transposed conv
// Block: 256 thr = 8 waves. Wave w: co-tile (w&3)*16, parity p = w>>2.
// Block covers out[b, 0..63, oz, oy, ox0..ox0+31].
// Grid rows staged into LDS once per block via GLOBAL_LOAD_ASYNC_TO_LDS_B128.
__global__ __launch_bounds__(256) void deconv_wmma_kernel(const float* __restrict__ grid,
                                                          const float* __restrict__ wb,
                                                          float* __restrict__ out) {
    __shared__ float ldsG[4 * ROW_F32];   // up to 4 (z,y) rows x 17 voxels x 64 ch
    __shared__ float ldsT[COUT * 33];     // store-transpose tile

    const int tid  = threadIdx.x;
    const int lane = tid & 31;
    const int w    = tid >> 5;
    const int h    = lane >> 4;          // 0/1 (half-wave)
    const int m16  = lane & 15;
    const int co0  = (w & 3) * 16;
    const int p    = w >> 2;             // 0 = even ox, 1 = odd ox

    const int xc = blockIdx.x % 3;
    const int oy = blockIdx.x / 3;
    const int oz = blockIdx.y;
    const int b  = blockIdx.z;
    const int ox0 = xc * 32;
    const int xb  = ox0 >> 1;

    // tap sets (kernel index, input coord); per dim: even o -> 1 tap, odd -> 2
    int zk[2], zz[2], nz;
    if (oz & 1) { nz = 2; zk[0] = 0; zz[0] = (oz + 1) >> 1; zk[1] = 2; zz[1] = (oz - 1) >> 1; }
    else        { nz = 1; zk[0] = 1; zz[0] = oz >> 1; }
    int yk[2], yy[2], ny;
    if (oy & 1) { ny = 2; yk[0] = 0; yy[0] = (oy + 1) >> 1; yk[1] = 2; yy[1] = (oy - 1) >> 1; }
    else        { ny = 1; yk[0] = 1; yy[0] = oy >> 1; }
    // x taps: parity p is wave-uniform; LDS x-offset delta per tap
    int xk[2], xdx[2], nx;
    if (p == 0) { nx = 1; xk[0] = 1; xdx[0] = 0; }
    else        { nx = 2; xk[0] = 0; xdx[0] = 1; xk[1] = 2; xdx[1] = 0; }

    // ---- stage nz*ny grid rows (x in [xb, xb+16], all 64 ch) into LDS ----
    const int nzy = nz * ny;
    for (int q = tid; q < nzy * ROW_F4; q += 256) {
        const int r  = q / ROW_F4;
        const int e  = q - r * ROW_F4;          // float4 index within row
        const int iz = r / ny;
        const int iy = r - iz * ny;
        const float* src = grid + ((((b * GD + zz[iz]) * GD + yy[iy]) * GD) + xb) * CIN + e * 4;
        const unsigned ldst = (unsigned)(unsigned long long)(&ldsG[r * ROW_F32 + e * 4]);
        asm volatile("global_load_async_to_lds_b128 %0, %1, off"
                     :: "v"(ldst), "v"(src) : "memory");
    }
    asm volatile("s_wait_asynccnt 0x0" ::: "memory");
    __syncthreads();

    // ---- WMMA K-loop: taps x 16 ci-chunks; A from global (L2), B from LDS ----
    v8f acc = {0.f, 0.f, 0.f, 0.f, 0.f, 0.f, 0.f, 0.f};

    for (int iz = 0; iz < nz; ++iz) {
        for (int iy = 0; iy < ny; ++iy) {
            const float* lrow = ldsG + (iz * ny + iy) * ROW_F32;
            for (int ix = 0; ix < nx; ++ix) {
                const int t3 = (zk[iz] * 3 + yk[iy]) * 3 + xk[ix];
                const float* lb    = lrow + (m16 + xdx[ix]) * CIN + 2 * h;        // B frag
                const float* wbase = wb + t3 * 4096 + h * 128 + (co0 + m16) * 2;  // A frag
#pragma unroll
                for (int cbi = 0; cbi < 16; ++cbi) {
                    v2f a  = *(const v2f*)(wbase + cbi * 256);
                    v2f bv = *(const v2f*)(lb + cbi * 4);
                    acc = __builtin_amdgcn_wmma_f32_16x16x4_f32(
                        false, a, false, bv, (short)0, acc, false, false);
                }
            }
        }
    }

    // D layout: lane -> column N=m16 (ox_local = p + 2*m16); acc[r] -> row co0 + r + 8*h
    {
        const int oxl = p + 2 * m16;
        float* lp = ldsT + (co0 + 8 * h) * 33 + oxl;
#pragma unroll
        for (int r = 0; r < 8; ++r) lp[r * 33] = acc[r];
    }
    __syncthreads();

    // coalesced non-temporal stores: 64 co rows x 32 consecutive ox
#pragma unroll
    for (int j = 0; j < 8; ++j) {
        int idx = j * 256 + tid;
        int co  = idx >> 5;
        int oxl = idx & 31;
        int ox  = ox0 + oxl;
        if (ox < OD) {
            int oidx = (((b * COUT + co) * OD + oz) * OD + oy) * OD + ox;
            __builtin_nontemporal_store(ldsT[co * 33 + oxl], out + oidx);
        }
    }
}

// ------------------------------------------------------------------- launch
extern "C" void kernel_launch(void* const* d_in, const int* in_sizes, int n_in,
                              void* d_out, int out_size, void* d_ws, size_t ws_size,
                              hipStream_t stream) {
    const float* features = (const float*)d_in[0];
    const int*   coors    = (const int*)d_in[1];
    const float* w1       = (const float*)d_in[3];
    float* out  = (float*)d_out;
    float* grid = (float*)d_ws;                 // 56.6 MB
    float* wb   = grid + GRID_ELEMS;            // +442 KB

    const int N = in_sizes[1] / 4;              // voxel count from coors

    // 1) zero the dense grid
    {
        int n4 = GRID_ELEMS / 4;
        zero_grid_kernel<<<(n4 + 255) / 256, 256, 0, stream>>>((float4*)grid, n4);
    }
    // 2) repack weights into WMMA A-fragment layout
    repack_w_kernel<<<(WB_ELEMS + 255) / 256, 256, 0, stream>>>(w1, wb);
    // 3) scatter-add features into grid (hardware f32 atomics, coalesced in C)
    {
        int total = N * CIN;
        scatter_kernel<<<(total + 255) / 256, 256, 0, stream>>>(features, coors, grid, total);
    }
    // 4) gather transposed-conv via v_wmma_f32_16x16x4_f32 + async LDS staging
    {
        dim3 g(3 * OD, OD, NB);                 // (285, 95, 2)
        deconv_wmma_kernel<<<g, 256, 0, stream>>>(grid, wb, out);
    }
}